// DissimilarityMixtureEncoderCov_64965675319308
// MI455X (gfx1250) — compile-verified
//
#include <hip/hip_runtime.h>
#include <math.h>

typedef float v2f __attribute__((ext_vector_type(2)));
typedef float v4f __attribute__((ext_vector_type(4)));
typedef float v8f __attribute__((ext_vector_type(8)));

#define ALPHA 10.0f
#define Bn 8192
#define Kn 128
#define Dn 128

// ---------------------------------------------------------------------------
// Kernel A: per mixture component k:
//   v[k][e]    = sum_d centers[k][d] * cov[k][d][e]   (= mu_k @ X_k)
//   covT[k][e][d] = cov[k][d][e]                      (transpose for b64 B-frags)
// ---------------------------------------------------------------------------
__global__ void dme_prep_kernel(const float* __restrict__ centers,
                                const float* __restrict__ cov,
                                float* __restrict__ v,
                                float* __restrict__ covT) {
    const int k = blockIdx.x;
    const int e = threadIdx.x;
    const float* covk  = cov  + (size_t)k * Dn * Dn;
    float*       covTk = covT + (size_t)k * Dn * Dn;
    const float* ck    = centers + (size_t)k * Dn;
    float acc = 0.0f;
    #pragma unroll 8
    for (int d = 0; d < Dn; ++d) {
        float val = covk[(size_t)d * Dn + e];   // coalesced across e
        acc = fmaf(ck[d], val, acc);
        covTk[(size_t)e * Dn + d] = val;        // transposed store
    }
    v[(size_t)k * Dn + e] = acc;
}

// ---------------------------------------------------------------------------
// Kernel B: D[b,k] = || x_b @ X_k - v_k ||^2 via V_WMMA_F32_16X16X4_F32.
//   block = 256 threads (8 waves); wave handles 16 rows of x, 32 values of k.
//   grid  = (Bn/128 row tiles, 4 k quarters)
// WMMA f32 16x16x4 fragment layouts (ISA 7.12.2):
//   A (16x4):  lane L: row m = L&15, g = L>>4; v2f = A[m][2g], A[m][2g+1]
//   B (4x16):  lane L: col n = L&15;           v2f = B[2g][n], B[2g+1][n]
//   C/D:       lane L: col n = L&15; c[r] = C[r + 8g][n]
// ---------------------------------------------------------------------------
__global__ void dme_maha_kernel(const float* __restrict__ x,
                                const float* __restrict__ covT,
                                const float* __restrict__ v,
                                float* __restrict__ out) {
    const int lane = threadIdx.x & 31;
    const int wave = threadIdx.x >> 5;
    const int n    = lane & 15;          // tile column (and A row)
    const int g    = lane >> 4;          // half-wave group
    const int m0   = blockIdx.x * 128 + wave * 16;   // first batch row of this wave
    const int k0   = blockIdx.y * 32;                // k quarter

    // Preload A fragments: x[m0+n][4t + 2g + {0,1}] for t = 0..31 (b64 loads).
    v2f a[32];
    const float* xrow = x + (size_t)(m0 + n) * Dn + 2 * g;
    #pragma unroll
    for (int t = 0; t < 32; ++t)
        a[t] = *(const v2f*)(xrow + 4 * t);

    #pragma unroll 1
    for (int k = k0; k < k0 + 32; ++k) {
        const float* covTk = covT + (size_t)k * Dn * Dn;
        float rowsum[8];
        #pragma unroll
        for (int r = 0; r < 8; ++r) rowsum[r] = 0.0f;

        #pragma unroll 1
        for (int et = 0; et < 8; ++et) {             // 16-wide e column tiles
            v8f acc0 = {0.f,0.f,0.f,0.f,0.f,0.f,0.f,0.f};
            v8f acc1 = {0.f,0.f,0.f,0.f,0.f,0.f,0.f,0.f};
            // B fragment: covT[k][et*16+n][4t + 2g + {0,1}] -> contiguous b64
            const float* bcol = covTk + (size_t)(et * 16 + n) * Dn + 2 * g;
            #pragma unroll
            for (int t = 0; t < 32; t += 2) {
                v2f bf0 = *(const v2f*)(bcol + 4 * t);
                v2f bf1 = *(const v2f*)(bcol + 4 * t + 4);
                acc0 = __builtin_amdgcn_wmma_f32_16x16x4_f32(
                           false, a[t],     false, bf0, (short)0, acc0, false, false);
                acc1 = __builtin_amdgcn_wmma_f32_16x16x4_f32(
                           false, a[t + 1], false, bf1, (short)0, acc1, false, false);
            }
            const float vv = v[(size_t)k * Dn + et * 16 + n];
            #pragma unroll
            for (int r = 0; r < 8; ++r) {
                float y = (acc0[r] + acc1[r]) - vv;   // Y[8g+r][e] - v_k[e]
                rowsum[r] = fmaf(y, y, rowsum[r]);
            }
        }

        // Reduce squared-norm partials across the 16 lanes of each half-group.
        #pragma unroll
        for (int r = 0; r < 8; ++r) {
            float s = rowsum[r];
            s += __shfl_xor(s, 1, 32);
            s += __shfl_xor(s, 2, 32);
            s += __shfl_xor(s, 4, 32);
            s += __shfl_xor(s, 8, 32);
            rowsum[r] = s;
        }
        // One writer lane per half-group stores the 8 row distances for this k.
        if (n == (k & 15)) {
            #pragma unroll
            for (int r = 0; r < 8; ++r)
                out[(size_t)(m0 + 8 * g + r) * Kn + k] = rowsum[r];
        }
    }
}

// ---------------------------------------------------------------------------
// Kernel C: in-place row softmax of (-ALPHA * D + log(relu(mixers))).
//   block = 256 threads (8 waves); one wave per batch row; 4 k per lane.
// ---------------------------------------------------------------------------
__global__ void dme_softmax_kernel(const float* __restrict__ mixers,
                                   float* __restrict__ out) {
    const int lane = threadIdx.x & 31;
    const int wave = threadIdx.x >> 5;
    const int row  = blockIdx.x * 8 + wave;
    float* rowp = out + (size_t)row * Kn;

    v4f d = *(const v4f*)(rowp + lane * 4);
    v4f m = *(const v4f*)(mixers + lane * 4);
    float lg[4];
    #pragma unroll
    for (int j = 0; j < 4; ++j) {
        float bias = logf(fmaxf(m[j], 0.0f));   // log(relu(mixers)), -inf if <= 0
        lg[j] = fmaf(-ALPHA, d[j], bias);
    }
    float mx = fmaxf(fmaxf(lg[0], lg[1]), fmaxf(lg[2], lg[3]));
    mx = fmaxf(mx, __shfl_xor(mx, 1, 32));
    mx = fmaxf(mx, __shfl_xor(mx, 2, 32));
    mx = fmaxf(mx, __shfl_xor(mx, 4, 32));
    mx = fmaxf(mx, __shfl_xor(mx, 8, 32));
    mx = fmaxf(mx, __shfl_xor(mx, 16, 32));
    float sum = 0.0f;
    #pragma unroll
    for (int j = 0; j < 4; ++j) { lg[j] = expf(lg[j] - mx); sum += lg[j]; }
    sum += __shfl_xor(sum, 1, 32);
    sum += __shfl_xor(sum, 2, 32);
    sum += __shfl_xor(sum, 4, 32);
    sum += __shfl_xor(sum, 8, 32);
    sum += __shfl_xor(sum, 16, 32);
    const float inv = 1.0f / sum;
    v4f o;
    #pragma unroll
    for (int j = 0; j < 4; ++j) o[j] = lg[j] * inv;
    *(v4f*)(rowp + lane * 4) = o;
}

// ---------------------------------------------------------------------------
extern "C" void kernel_launch(void* const* d_in, const int* in_sizes, int n_in,
                              void* d_out, int out_size, void* d_ws, size_t ws_size,
                              hipStream_t stream) {
    const float* x       = (const float*)d_in[0];   // (8192,128)
    const float* centers = (const float*)d_in[1];   // (128,128)
    const float* cov     = (const float*)d_in[2];   // (128,128,128)
    const float* mixers  = (const float*)d_in[3];   // (1,128)
    float* out = (float*)d_out;                     // (8192,128)

    // Workspace: v (K*D floats) then covT (K*D*D floats) ~ 8.06 MB total.
    float* v    = (float*)d_ws;
    float* covT = (float*)d_ws + (size_t)Kn * Dn;

    dme_prep_kernel<<<dim3(Kn), dim3(Dn), 0, stream>>>(centers, cov, v, covT);
    dme_maha_kernel<<<dim3(Bn / 128, 4), dim3(256), 0, stream>>>(x, covT, v, out);
    dme_softmax_kernel<<<dim3(Bn / 8), dim3(256), 0, stream>>>(mixers, out);
}